// Mask2FormerForSegmentation_46291157516869
// MI455X (gfx1250) — compile-verified
//
#include <hip/hip_runtime.h>

// Problem constants (match the reference setup exactly).
#define B_    8
#define Q_    100
#define C_    19     // real classes (C+1 logits per query, last sliced off)
#define MH_   128
#define MW_   128
#define OUT_  512
#define TILE_ROWS 32            // output rows per block (16 tiles cover 512)
#define SEG_ROWS  10            // input rows feeding one tile (r0/4-1 .. r0/4+8)
#define TPB   256
#define ELEMS (SEG_ROWS * MW_)  // 1280 floats per staged tile-slab
#define EPT   (ELEMS / TPB)     // 5 elements per thread
#define NBUF  3                 // triple buffer: 2 iters between consume & overwrite

typedef float v4f __attribute__((ext_vector_type(4)));  // native vector: NT-storable

// CDNA5 async global->LDS copy (ASYNCcnt path). First operand: VGPR holding the
// LDS byte offset (low 32 bits of the generic pointer to __shared__), second:
// 64-bit global address in a VGPR pair, saddr disabled.
__device__ __forceinline__ void async_ld_b32(const float* gsrc, unsigned lds_off) {
  asm volatile("global_load_async_to_lds_b32 %0, %1, off"
               :: "v"(lds_off), "v"(gsrc)
               : "memory");
}
__device__ __forceinline__ void wait_async_le5() {
  asm volatile("s_wait_asynccnt 0x5" ::: "memory");
}
__device__ __forceinline__ void wait_async_le0() {
  asm volatile("s_wait_asynccnt 0x0" ::: "memory");
}

__global__ __launch_bounds__(TPB)
void m2f_fused_kernel(const float* __restrict__ cls,
                      const float* __restrict__ masks,
                      float* __restrict__ out) {
  const int t  = threadIdx.x;
  const int ch = blockIdx.y;            // ch = b*C + c  (== output channel index)
  const int b  = ch / C_;
  const int c  = ch % C_;
  const int r0 = blockIdx.x * TILE_ROWS;   // first output row of this tile
  const int ib = (r0 >> 2) - 1;            // first (unclamped) input row needed

  __shared__ int   s_pred[Q_];
  __shared__ int   s_ql[Q_];
  __shared__ int   s_nq;
  __shared__ float seg[ELEMS];                 // class-max tile (10 x 128)
  __shared__ float stage[NBUF][ELEMS];         // async staging buffers

  // ---- per-query argmax over the 19 real classes (first-max tiebreak) ----
  if (t < Q_) {
    const float* p = cls + (size_t)(b * Q_ + t) * (C_ + 1);
    float best = p[0]; int bi = 0;
#pragma unroll
    for (int j = 1; j < C_; ++j) { float v = p[j]; if (v > best) { best = v; bi = j; } }
    s_pred[t] = bi;
  }
  __syncthreads();
  if (t == 0) {
    int n = 0;
    for (int q = 0; q < Q_; ++q) if (s_pred[q] == c) s_ql[n++] = q;
    s_nq = n;
  }
  __syncthreads();
  const int nq = s_nq;

  // ---- per-thread element offsets inside one 128x128 mask (rows edge-clamped) ----
  int eoff[EPT];
#pragma unroll
  for (int j = 0; j < EPT; ++j) {
    int e    = t + j * TPB;        // 0..1279
    int kr   = e >> 7;             // LDS row 0..9
    int col  = e & (MW_ - 1);
    int grow = ib + kr;
    grow = grow < 0 ? 0 : (grow > MH_ - 1 ? MH_ - 1 : grow);
    eoff[j] = grow * MW_ + col;
  }
  const float* mb = masks + (size_t)b * Q_ * (MH_ * MW_);
  const unsigned stage0 = (unsigned)(size_t)&stage[0][0];  // LDS byte offset

  float acc[EPT];
#pragma unroll
  for (int j = 0; j < EPT; ++j) acc[j] = -3.402823466e+38f;

  if (nq > 0) {
    { // prologue: prefetch query 0 into buffer 0
      const float* gq = mb + (size_t)s_ql[0] * (MH_ * MW_);
#pragma unroll
      for (int j = 0; j < EPT; ++j)
        async_ld_b32(gq + eoff[j], stage0 + (unsigned)((t + j * TPB) * 4));
    }
    for (int i = 0; i < nq; ++i) {
      const int cur = i % NBUF;
      if (i + 1 < nq) {                       // prefetch next query, then wait for cur
        const int nxt = (i + 1) % NBUF;
        const float* gq = mb + (size_t)s_ql[i + 1] * (MH_ * MW_);
#pragma unroll
        for (int j = 0; j < EPT; ++j)
          async_ld_b32(gq + eoff[j],
                       stage0 + (unsigned)((nxt * ELEMS + t + j * TPB) * 4));
        wait_async_le5();                     // oldest 5 (cur buffer) retired
      } else {
        wait_async_le0();                     // drain
      }
      volatile const float* sb = &stage[cur][0];
#pragma unroll
      for (int j = 0; j < EPT; ++j)
        acc[j] = fmaxf(acc[j], sb[t + j * TPB]);
    }
  }
#pragma unroll
  for (int j = 0; j < EPT; ++j)
    seg[t + j * TPB] = (nq > 0) ? acc[j] : 0.0f;   // empty class -> zeros
  __syncthreads();

  // ---- half-pixel bilinear 4x upsample: x = o/4 - 0.375, 4 fixed phases ----
  const float fyc[4] = {0.625f, 0.875f, 0.125f, 0.375f};
#pragma unroll 4
  for (int it = 0; it < (TILE_ROWS * (OUT_ / 4)) / TPB; ++it) {  // 4096/256 = 16
    int qd = t + it * TPB;          // quad index: 32 rows x 128 quads
    int rl = qd >> 7;               // local output row 0..31
    int qc = qd & 127;              // quad column -> ocol = 4*qc
    int r  = r0 + rl;
    int rm = r & 3;
    int iy = (r >> 2) + ((rm >= 2) ? 0 : -1);
    int k0 = iy - ib;               // 0..8 (LDS rows already edge-clamped)
    float fy = fyc[rm];
    int cm1 = qc - 1; cm1 = cm1 < 0 ? 0 : cm1;
    int cp1 = qc + 1; cp1 = cp1 > MW_ - 1 ? MW_ - 1 : cp1;
    const float* s0 = &seg[k0 * MW_];
    const float* s1 = &seg[(k0 + 1) * MW_];
    float a = s0[cm1] + fy * (s1[cm1] - s0[cm1]);
    float m = s0[qc ] + fy * (s1[qc ] - s0[qc ]);
    float p = s0[cp1] + fy * (s1[cp1] - s0[cp1]);
    v4f o;
    o.x = 0.375f * a + 0.625f * m;
    o.y = 0.125f * a + 0.875f * m;
    o.z = 0.875f * m + 0.125f * p;
    o.w = 0.625f * m + 0.375f * p;
    // Output is written once and never re-read: stream it past L2 (NT hint)
    // so the 52 MB of mask data stays L2-resident for cross-tile re-reads.
    v4f* dst = (v4f*)(out + ((size_t)ch * OUT_ + r) * OUT_ + (size_t)qc * 4);
    __builtin_nontemporal_store(o, dst);
  }
}

extern "C" void kernel_launch(void* const* d_in, const int* in_sizes, int n_in,
                              void* d_out, int out_size, void* d_ws, size_t ws_size,
                              hipStream_t stream) {
  (void)in_sizes; (void)n_in; (void)d_ws; (void)ws_size; (void)out_size;
  const float* cls   = (const float*)d_in[0];   // [B, Q, C+1] f32
  const float* masks = (const float*)d_in[1];   // [B, Q, 128, 128] f32
  // d_in[2] is image_size (=512), baked in as OUT_.
  float* out = (float*)d_out;                   // [B, C, 512, 512] f32
  dim3 grid(OUT_ / TILE_ROWS, B_ * C_);         // 16 x 152
  m2f_fused_kernel<<<grid, TPB, 0, stream>>>(cls, masks, out);
}